// Binarization_23630910063173
// MI455X (gfx1250) — compile-verified
//
#include <hip/hip_runtime.h>
#include <hip/hip_bf16.h>
#include <stdint.h>

// ---------------------------------------------------------------------------
// Order-2 residual binarization + 15 alternating refinement iterations.
// One workgroup (1024 threads = 32 wave32 waves) per row of 11008 f32.
// x row staged into LDS once via the Tensor Data Mover (CDNA5 TDM); mask
// byte loads are issued by all waves while the DMA is in flight. Row data is
// then held register-resident (11 elems/thread) for all ~19 fused sweeps.
// HBM traffic: one read of x+mask, one write of out  (~370 MB ~ 16us floor).
// ---------------------------------------------------------------------------

#define COLS   11008
#define BLOCK  1024
#define WAVES  32          // wave32
#define NPT    11          // ceil(COLS / BLOCK)

typedef unsigned int u32x4 __attribute__((ext_vector_type(4)));
typedef int          i32x4 __attribute__((ext_vector_type(4)));
typedef int          i32x8 __attribute__((ext_vector_type(8)));

__device__ __forceinline__ float wave_sum(float v) {
#pragma unroll
  for (int off = 16; off > 0; off >>= 1) v += __shfl_xor(v, off, 32);
  return v;
}

// Two-stage block reduction: wave32 shuffle tree, then wave 0 combines the
// 32 wave partials. Results broadcast to every thread through `fin`.
template <int NP>
__device__ __forceinline__ void block_reduce(float (&v)[NP], float (*red)[8],
                                             float* fin, int lane, int wave) {
#pragma unroll
  for (int q = 0; q < NP; ++q) {
    float s = wave_sum(v[q]);
    if (lane == 0) red[wave][q] = s;
  }
  __syncthreads();
  if (wave == 0) {
#pragma unroll
    for (int q = 0; q < NP; ++q) {
      float s = wave_sum(red[lane][q]);
      if (lane == 0) fin[q] = s;
    }
  }
  __syncthreads();
#pragma unroll
  for (int q = 0; q < NP; ++q) v[q] = fin[q];
}

__global__ __launch_bounds__(BLOCK) void
Binarization_23630910063173_kernel(const float* __restrict__ x,
                                   const unsigned char* __restrict__ mask,
                                   const int* __restrict__ num_iters_p,
                                   float* __restrict__ out) {
  __shared__ float xrow[COLS];          // 44 KB row staging (TDM destination)
  __shared__ float red[WAVES][8];
  __shared__ float fin[8];

  const int row  = blockIdx.x;
  const int tid  = threadIdx.x;
  const int lane = tid & 31;
  const int wave = tid >> 5;
  const int num_iters = num_iters_p[0];

  // ---- Issue the TDM DMA for this row (wave 0, no wait yet) ---------------
  const uint64_t gaddr = (uint64_t)(uintptr_t)(x + (size_t)row * COLS);
#if __has_builtin(__builtin_amdgcn_tensor_load_to_lds)
  if (wave == 0) {
    const uint32_t lds_off = (uint32_t)(uintptr_t)&xrow[0];  // low 32b = LDS offset
    u32x4 g0;
    g0[0] = 1u;                                        // count=1, user mode
    g0[1] = lds_off;                                   // lds_addr
    g0[2] = (uint32_t)(gaddr & 0xFFFFFFFFu);           // global_addr[31:0]
    g0[3] = (uint32_t)((gaddr >> 32) & 0x01FFFFFFu)    // global_addr[56:32]
            | (2u << 30);                              // type = 2 ("image")
    i32x8 g1;
    g1[0] = (int)(2u << 16);                           // data_size = 4 bytes
    g1[1] = (int)((uint32_t)(COLS & 0xFFFF) << 16);    // tensor_dim0 lo16
    g1[2] = (int)(((uint32_t)COLS >> 16) | (1u << 16));// dim0 hi16 | tensor_dim1=1
    g1[3] = (int)((uint32_t)(COLS & 0xFFFF) << 16);    // tile_dim0 = COLS
    g1[4] = 1;                                         // tile_dim1=1, tile_dim2=0
    g1[5] = COLS;                                      // tensor_dim0_stride lo32
    g1[6] = 0;
    g1[7] = 0;
    i32x4 gz = {0, 0, 0, 0};
#if defined(__clang_major__) && (__clang_major__ >= 23)
    i32x8 gz8 = {0, 0, 0, 0, 0, 0, 0, 0};
    __builtin_amdgcn_tensor_load_to_lds(g0, g1, gz, gz, gz8, 0);
#else
    __builtin_amdgcn_tensor_load_to_lds(g0, g1, gz, gz, 0);
#endif
  }
#else
  for (int j = tid; j < COLS; j += BLOCK) xrow[j] = x[(size_t)row * COLS + j];
#endif

  // ---- Overlap: all waves load their mask bytes while the DMA runs --------
  unsigned mbits = 0;
#pragma unroll
  for (int k = 0; k < NPT; ++k) {
    const int j = tid + k * BLOCK;
    if (j < COLS && mask[(size_t)row * COLS + j] != 0) mbits |= (1u << k);
  }

  // ---- Drain the tensor DMA, make LDS row visible to all waves ------------
#if __has_builtin(__builtin_amdgcn_tensor_load_to_lds)
  if (wave == 0) __builtin_amdgcn_s_wait_tensorcnt(0);
#endif
  __syncthreads();

  // ---- Pull this thread's 11 elements into registers ----------------------
  float xm[NPT], b0[NPT], b1[NPT];
#pragma unroll
  for (int k = 0; k < NPT; ++k) {
    const int j = tid + k * BLOCK;
    const bool mv = ((mbits >> k) & 1u) != 0;
    xm[k] = (mv && j < COLS) ? xrow[j] : 0.0f;   // x * maskf
    b0[k] = 0.0f; b1[k] = 0.0f;
  }

  // ---- Phase 1, pass 1: S_x, cnt --> mu1 ----------------------------------
  float r2[2] = {0.0f, 0.0f};
#pragma unroll
  for (int k = 0; k < NPT; ++k) {
    const float m = ((mbits >> k) & 1u) ? 1.0f : 0.0f;
    r2[0] += xm[k];
    r2[1] += m;
  }
  block_reduce(r2, red, fin, lane, wave);
  const float S_x = r2[0], cnt = r2[1];
  const float inv_cnt = 1.0f / fmaxf(cnt, 1.0f);
  const bool  has = (cnt > 0.0f);
  const float mu1 = has ? S_x * inv_cnt : 0.0f;

  // alpha1 = masked mean |xm - mu1|
  float r1[1] = {0.0f};
#pragma unroll
  for (int k = 0; k < NPT; ++k) {
    const float m = ((mbits >> k) & 1u) ? 1.0f : 0.0f;
    r1[0] += m * fabsf(xm[k] - mu1);
  }
  block_reduce(r1, red, fin, lane, wave);
  const float alpha1 = has ? r1[0] * inv_cnt : 0.0f;

  // b0 = sign(masked centered)
#pragma unroll
  for (int k = 0; k < NPT; ++k) {
    const float c = ((mbits >> k) & 1u) ? (xm[k] - mu1) : 0.0f;
    b0[k] = (c > 0.0f) ? 1.0f : ((c < 0.0f) ? -1.0f : 0.0f);
  }

  // ---- Phase 1, pass 2: mu2, alpha2, b1 -----------------------------------
  r1[0] = 0.0f;
#pragma unroll
  for (int k = 0; k < NPT; ++k) {
    const float m = ((mbits >> k) & 1u) ? 1.0f : 0.0f;
    r1[0] += m * (xm[k] - (b0[k] * alpha1 + mu1));     // residual2
  }
  block_reduce(r1, red, fin, lane, wave);
  const float mu2 = has ? r1[0] * inv_cnt : 0.0f;
  float rmean = mu1 + mu2;

  r1[0] = 0.0f;
#pragma unroll
  for (int k = 0; k < NPT; ++k) {
    const float m = ((mbits >> k) & 1u) ? 1.0f : 0.0f;
    r1[0] += m * fabsf(xm[k] - b0[k] * alpha1 - mu1 - mu2);
  }
  block_reduce(r1, red, fin, lane, wave);
  const float alpha2 = has ? r1[0] * inv_cnt : 0.0f;

#pragma unroll
  for (int k = 0; k < NPT; ++k) {
    const float c = ((mbits >> k) & 1u) ? (xm[k] - b0[k] * alpha1 - rmean) : 0.0f;
    b1[k] = (c > 0.0f) ? 1.0f : ((c < 0.0f) ? -1.0f : 0.0f);
  }

  float a0 = alpha1, a1 = alpha2;

  // ---- Alternating refinement: fused (reduce -> scalars -> argmin) --------
  for (int it = 0; it < num_iters; ++it) {
    float r7[7] = {0, 0, 0, 0, 0, 0, 0};
#pragma unroll
    for (int k = 0; k < NPT; ++k) {
      const float m  = ((mbits >> k) & 1u) ? 1.0f : 0.0f;
      const float B0 = m * b0[k], B1 = m * b1[k];
      r7[0] += B0;              // sum b0
      r7[1] += B1;              // sum b1
      r7[2] += B0 * B1;         // sum b0*b1
      r7[3] += B0 * xm[k];      // sum b0*x
      r7[4] += B1 * xm[k];      // sum b1*x
      r7[5] += B0 * B0;         // sum b0^2  (phase-1 signs can be 0)
      r7[6] += B1 * B1;         // sum b1^2
    }
    block_reduce(r7, red, fin, lane, wave);
    const float Sb0 = r7[0], Sb1 = r7[1], Sb01 = r7[2];
    const float T0 = r7[3], T1 = r7[4], Q0 = r7[5], Q1 = r7[6];

    const float mu = has ? (S_x - a0 * Sb0 - a1 * Sb1 - rmean * cnt) * inv_cnt
                         : 0.0f;
    rmean += mu;
    a0 = (T0 - rmean * Sb0 - a1 * Sb01) / (Q0 + 1e-8f);
    a1 = (T1 - rmean * Sb1 - a0 * Sb01) / (Q1 + 1e-8f);

    // nearest of {-a0-a1, -a0+a1, a0-a1, a0+a1}; first-min tie-break (argmin)
#pragma unroll
    for (int k = 0; k < NPT; ++k) {
      const float t  = xm[k] - rmean;
      const float d0 = fabsf(t + a0 + a1);
      const float d1 = fabsf(t + a0 - a1);
      const float d2 = fabsf(t - a0 + a1);
      const float d3 = fabsf(t - a0 - a1);
      int idx = 0; float best = d0;
      if (d1 < best) { best = d1; idx = 1; }
      if (d2 < best) { best = d2; idx = 2; }
      if (d3 < best) { best = d3; idx = 3; }
      b0[k] = (idx <= 1)       ? -1.0f : 1.0f;
      b1[k] = ((idx & 1) == 0) ? -1.0f : 1.0f;
    }
  }

  // ---- sum_alt = (a0*b0 + a1*b1 + rmean) * maskf --------------------------
#pragma unroll
  for (int k = 0; k < NPT; ++k) {
    const int j = tid + k * BLOCK;
    if (j < COLS) {
      const float m = ((mbits >> k) & 1u) ? 1.0f : 0.0f;
      out[(size_t)row * COLS + j] = m * (a0 * b0[k] + a1 * b1[k] + rmean);
    }
  }
}

extern "C" void kernel_launch(void* const* d_in, const int* in_sizes, int n_in,
                              void* d_out, int out_size, void* d_ws, size_t ws_size,
                              hipStream_t stream) {
  const float*         x    = (const float*)d_in[0];
  const unsigned char* mask = (const unsigned char*)d_in[1];  // jax bool: 1 byte/elem
  // d_in[2] = order (== 2; algorithm structure is hardcoded for order 2)
  const int*           nit  = (const int*)d_in[3];
  float*               out  = (float*)d_out;

  const int rows = in_sizes[0] / COLS;   // 4096
  hipLaunchKernelGGL(Binarization_23630910063173_kernel,
                     dim3(rows), dim3(BLOCK), 0, stream, x, mask, nit, out);
}